// HardL1ACELoss_84404697301125
// MI455X (gfx1250) — compile-verified
//
#include <hip/hip_runtime.h>
#include <stdint.h>

#define NBINS 20
#define NCOPY 16          // replicated LDS histograms per block (one per 16 lanes)
#define COPY_STRIDE 68    // 64 + 4 -> rotates bank assignment by 4 banks per copy
#define DEPTH 4           // async pipeline depth (tiles in flight per thread)
#define TPB 256
#define MAX_BLOCKS 2048

typedef __attribute__((ext_vector_type(2))) float v2f;
typedef __attribute__((ext_vector_type(8))) float v8f;

// Issue one 16-byte async global->LDS copy (ASYNCcnt-tracked, CDNA5 path).
__device__ __forceinline__ void async_load16(unsigned lds_byte_addr, const void* gaddr) {
    asm volatile("global_load_async_to_lds_b128 %0, %1, off"
                 :: "v"(lds_byte_addr), "v"(gaddr)
                 : "memory");
}

__global__ void ace_ws_init_kernel(float* ws) {
    if (threadIdx.x < 64) ws[threadIdx.x] = 0.0f;
}

__global__ void __launch_bounds__(TPB)
ace_hist_kernel(const float* __restrict__ preds,
                const float* __restrict__ targs,
                float* __restrict__ ws,
                long long n4)   // number of float4 tiles
{
    // Staging: DEPTH slots x 256 threads x 32B (16B preds + 16B targets)
    __shared__ float stage[DEPTH * TPB * 8];
    __shared__ float hist[NCOPY * COPY_STRIDE];

    const int tid = threadIdx.x;
    for (int i = tid; i < NCOPY * COPY_STRIDE; i += TPB) hist[i] = 0.0f;
    __syncthreads();

    // LDS flat address: low 32 bits of the generic pointer are the LDS byte offset.
    const unsigned stage_base   = (unsigned)(size_t)(void*)stage;
    const unsigned my_slot_base = stage_base + (unsigned)tid * 32u;

    const long long gstride    = (long long)gridDim.x * TPB;
    const long long tile0      = (long long)blockIdx.x * TPB + tid;
    const long long blockTile0 = (long long)blockIdx.x * TPB;   // block-uniform
    const long long numIters   = (blockTile0 < n4)
                               ? (n4 - blockTile0 + gstride - 1) / gstride : 0;

    // Prologue: fill the pipeline (issue conditions are block-uniform; per-lane
    // tails are clamped to a safe address and masked at consume time).
    for (int d = 0; d < DEPTH; ++d) {
        if ((long long)d < numIters) {
            long long g  = tile0 + (long long)d * gstride;
            long long gc = (g < n4) ? g : (n4 - 1);
            unsigned  lb = my_slot_base + (unsigned)d * (TPB * 32u);
            async_load16(lb,       preds + 4 * gc);
            async_load16(lb + 16u, targs + 4 * gc);
        }
    }

    const int copy = tid >> 4;   // 16 lanes share one histogram copy
    float* hc = &hist[copy * COPY_STRIDE];

    for (long long it = 0; it < numIters; ++it) {
        const int slot = (int)(it & (DEPTH - 1));

        // Wait until this tile's 2 async ops are done (in-order completion).
        // Outstanding beyond tile 'it' = 2 * min(DEPTH-1, numIters-1-it).
        long long beyond = numIters - 1 - it;
        if (beyond > DEPTH - 1) beyond = DEPTH - 1;
        switch ((int)beyond) {
            case 3:  asm volatile("s_wait_asynccnt 6" ::: "memory"); break;
            case 2:  asm volatile("s_wait_asynccnt 4" ::: "memory"); break;
            case 1:  asm volatile("s_wait_asynccnt 2" ::: "memory"); break;
            default: asm volatile("s_wait_asynccnt 0" ::: "memory"); break;
        }

        const long long g = tile0 + it * gstride;
        const bool valid  = (g < n4);
        const float4 p4 = *(const float4*)&stage[(slot * TPB + tid) * 8];
        const float4 t4 = *(const float4*)&stage[(slot * TPB + tid) * 8 + 4];

        if (valid) {
#pragma unroll
            for (int e = 0; e < 4; ++e) {
                const float p = (&p4.x)[e];
                const float t = (&t4.x)[e];
                const float w = (p >= 0.0f && p < 1.0f) ? 1.0f : 0.0f;
                int idx = (int)floorf(p * (float)NBINS);
                idx = idx < 0 ? 0 : (idx > NBINS - 1 ? NBINS - 1 : idx);
                atomicAdd(&hc[idx],             w);      // ds_add_f32 (no return)
                atomicAdd(&hc[NBINS + idx],     p * w);
                atomicAdd(&hc[2 * NBINS + idx], t * w);
            }
        }

        // Refill this slot with tile it+DEPTH.
        if (it + DEPTH < numIters) {
            long long gg = tile0 + (it + DEPTH) * gstride;
            long long gc = (gg < n4) ? gg : (n4 - 1);
            unsigned  lb = my_slot_base + (unsigned)slot * (TPB * 32u);
            async_load16(lb,       preds + 4 * gc);
            async_load16(lb + 16u, targs + 4 * gc);
        }
    }
    asm volatile("s_wait_asynccnt 0" ::: "memory");
    __syncthreads();

    // Reduce the 16 histogram copies. out[s] = sum_c hist[c][s], s in 0..59.
    if (tid < 32) {
#if __has_builtin(__builtin_amdgcn_wmma_f32_16x16x4_f32)
        // D += ones(16x4) x B(4x16): every D[m,n] = sum_k B[k,n]. A = all-ones
        // makes the K-slot lane mapping irrelevant; each lane loads 2 distinct
        // copies so the 4 copies of column n are covered exactly once.
        const int half = tid >> 4;     // which K half this lane feeds
        const int col  = tid & 15;     // column n
        v2f a; a.x = 1.0f; a.y = 1.0f;
#pragma unroll
        for (int j = 0; j < 4; ++j) {  // slot group: s = 16j + col
            v8f acc = {};
            const int s = 16 * j + col;
#pragma unroll
            for (int gq = 0; gq < 4; ++gq) {   // copy group of 4
                const int c0 = 4 * gq + 2 * half;
                v2f b;
                b.x = hist[(c0 + 0) * COPY_STRIDE + s];
                b.y = hist[(c0 + 1) * COPY_STRIDE + s];
                acc = __builtin_amdgcn_wmma_f32_16x16x4_f32(
                          false, a, false, b, (short)0, acc, false, false);
            }
            if (half == 0 && s < 3 * NBINS) atomicAdd(&ws[s], acc[0]);
        }
#else
        // Fallback: scalar copy reduction.
        for (int s = tid; s < 3 * NBINS; s += 32) {
            float acc = 0.0f;
#pragma unroll
            for (int c = 0; c < NCOPY; ++c) acc += hist[c * COPY_STRIDE + s];
            atomicAdd(&ws[s], acc);
        }
#endif
    }
}

__global__ void ace_finalize_kernel(const float* __restrict__ ws,
                                    const float* __restrict__ preds,
                                    const float* __restrict__ targs,
                                    long long n, long long ntail,
                                    float* __restrict__ out)
{
    if (threadIdx.x != 0 || blockIdx.x != 0) return;
    float cnt[NBINS], sp[NBINS], st[NBINS];
    for (int b = 0; b < NBINS; ++b) {
        cnt[b] = ws[b];
        sp[b]  = ws[NBINS + b];
        st[b]  = ws[2 * NBINS + b];
    }
    for (long long i = n - ntail; i < n; ++i) {   // scalar tail (n % 4 elements)
        const float p = preds[i], t = targs[i];
        const float w = (p >= 0.0f && p < 1.0f) ? 1.0f : 0.0f;
        int idx = (int)floorf(p * (float)NBINS);
        idx = idx < 0 ? 0 : (idx > NBINS - 1 ? NBINS - 1 : idx);
        cnt[idx] += w; sp[idx] += p * w; st[idx] += t * w;
    }
    float acesum = 0.0f, nvalid = 0.0f;
    for (int b = 0; b < NBINS; ++b) {
        if (cnt[b] > 0.0f) {
            acesum += fabsf(sp[b] - st[b]) / cnt[b];  // |sp/c - st/c| = |sp-st|/c
            nvalid += 1.0f;
        }
    }
    out[0] = acesum / fmaxf(nvalid, 1.0f);
}

extern "C" void kernel_launch(void* const* d_in, const int* in_sizes, int n_in,
                              void* d_out, int out_size, void* d_ws, size_t ws_size,
                              hipStream_t stream) {
    (void)n_in; (void)out_size; (void)ws_size;
    const float* preds = (const float*)d_in[0];
    const float* targs = (const float*)d_in[1];
    float* ws  = (float*)d_ws;
    float* out = (float*)d_out;

    const long long n     = (long long)in_sizes[0];
    const long long n4    = n / 4;
    const long long ntail = n - n4 * 4;

    ace_ws_init_kernel<<<1, 64, 0, stream>>>(ws);

    long long want = (n4 + TPB - 1) / TPB;
    int blocks = (int)(want < MAX_BLOCKS ? want : MAX_BLOCKS);
    if (blocks < 1) blocks = 1;
    ace_hist_kernel<<<blocks, TPB, 0, stream>>>(preds, targs, ws, n4);

    ace_finalize_kernel<<<1, 32, 0, stream>>>(ws, preds, targs, n, ntail, out);
}